// SingleHop_13125420057191
// MI455X (gfx1250) — compile-verified
//
#include <hip/hip_runtime.h>
#include <math.h>

// Problem constants (match reference)
#define B_      64
#define N_      4096
#define ENC     512
#define D       128
#define CHUNK_N 256
#define CHUNKS  (N_ / CHUNK_N)   // 16
#define NEG_INF (-1.0e30f)
#define EPS_    (1e-12f)

typedef float v2f __attribute__((ext_vector_type(2)));
typedef float v8f __attribute__((ext_vector_type(8)));
typedef int   v4i __attribute__((ext_vector_type(4)));

#define AS1 __attribute__((address_space(1)))
#define AS3 __attribute__((address_space(3)))

// Workspace layout (in floats):
//   proj : [B][D]
//   m    : [B][CHUNKS]
//   Z    : [B][CHUNKS]
//   p    : [B][CHUNKS][D]
#define PROJ_OFF 0
#define PROJ_SZ  (B_ * D)
#define M_OFF    (PROJ_OFF + PROJ_SZ)
#define M_SZ     (B_ * CHUNKS)
#define Z_OFF    (M_OFF + M_SZ)
#define Z_SZ     (B_ * CHUNKS)
#define P_OFF    (Z_OFF + Z_SZ)

// ---------------------------------------------------------------------------
// Kernel 0: proj_q[b][d] = sum_e vecQ[b][e] * Wq[d][e] + bq[d]
// ---------------------------------------------------------------------------
__global__ __launch_bounds__(128)
void proj_kernel(const float* __restrict__ vecQ,
                 const float* __restrict__ Wq,
                 const float* __restrict__ bq,
                 float* __restrict__ proj) {
  __shared__ float qLds[ENC];
  const int b = blockIdx.x;
  const int t = threadIdx.x;               // t = d in [0,128)
  const float* vq = vecQ + (size_t)b * ENC;
  for (int i = t; i < ENC; i += 128) qLds[i] = vq[i];
  __syncthreads();
  const float* row = Wq + (size_t)t * ENC;
  float acc = 0.f;
  for (int e = 0; e < ENC; e += 4) {
    float4 w = *(const float4*)(row + e);
    acc += w.x * qLds[e]     + w.y * qLds[e + 1]
         + w.z * qLds[e + 2] + w.w * qLds[e + 3];
  }
  proj[(size_t)b * D + t] = acc + bq[t];
}

// ---------------------------------------------------------------------------
// Kernel 1: per (batch, chunk) compute online-softmax partials.
//   Single global read of kb: chunk staged in LDS (128 KB) via async
//   global->LDS (VGPR-bypassing, ASYNCcnt-tracked) when available,
//   scores from LDS, partial weighted sum via V_WMMA_F32_16X16X4_F32.
// ---------------------------------------------------------------------------
__global__ __launch_bounds__(256)
void att_partial_kernel(const float* __restrict__ kb,
                        const int*   __restrict__ objNum,
                        const float* __restrict__ Wa,
                        const float* __restrict__ ba,
                        const float* __restrict__ proj,
                        float* __restrict__ ws_m,
                        float* __restrict__ ws_z,
                        float* __restrict__ ws_p) {
  __shared__ float tile[CHUNK_N * D];   // 128 KB (fits 320 KB WGP LDS)
  __shared__ float qv[D];               // proj_q
  __shared__ float qw[D];               // proj_q * Wa
  __shared__ float eBuf[CHUNK_N];
  __shared__ float red[8];

  const int chunk = blockIdx.x;
  const int b     = blockIdx.y;
  const int t     = threadIdx.x;
  const int lane  = t & 31;
  const int wave  = t >> 5;

  if (t < D) {
    float q = proj[(size_t)b * D + t];
    qv[t] = q;
    qw[t] = q * Wa[t];
  }

  // Chunk fill: 256 slots x 128 dims = 8192 float4s, 32 per thread.
  const float*  src  = kb + ((size_t)b * N_ + (size_t)chunk * CHUNK_N) * D;
  const float4* src4 = (const float4*)src;
  float4*       dst4 = (float4*)tile;
#if defined(__has_builtin) && __has_builtin(__builtin_amdgcn_global_load_async_to_lds_b128)
  {
    // Direct global->LDS DMA path (gfx1250): no VGPR round-trip.
    AS1 v4i* g = (AS1 v4i*)src4;
    AS3 v4i* l = (AS3 v4i*)dst4;
    for (int i = t; i < (CHUNK_N * D) / 4; i += 256) {
      __builtin_amdgcn_global_load_async_to_lds_b128(g + i, l + i, 0, 0);
    }
#if __has_builtin(__builtin_amdgcn_s_wait_asynccnt)
    __builtin_amdgcn_s_wait_asynccnt(0);
#else
    asm volatile("s_wait_asynccnt 0" ::: "memory");
#endif
  }
#else
  for (int i = t; i < (CHUNK_N * D) / 4; i += 256) dst4[i] = src4[i];
#endif
  __syncthreads();

  // One slot per thread: s = (sum kb*q*Wa) / max(||kb*q||, eps) + ba
  const int n = t;
  const float* rowp = tile + n * D;
  float dot1 = 0.f, dot2 = 0.f;
  for (int d = 0; d < D; d += 4) {
    float4 k4 = *(const float4*)(rowp + d);
    float i0 = k4.x * qv[d],     i1 = k4.y * qv[d + 1];
    float i2 = k4.z * qv[d + 2], i3 = k4.w * qv[d + 3];
    dot2 += i0 * i0 + i1 * i1 + i2 * i2 + i3 * i3;
    dot1 += k4.x * qw[d]     + k4.y * qw[d + 1]
          + k4.z * qw[d + 2] + k4.w * qw[d + 3];
  }
  float s = dot1 / fmaxf(sqrtf(dot2), EPS_) + ba[0];
  const int gn = chunk * CHUNK_N + n;
  if (gn >= objNum[b]) s = NEG_INF;

  // Chunk max (wave32 shuffle tree + cross-wave via LDS)
  float m = s;
  for (int off = 16; off > 0; off >>= 1) m = fmaxf(m, __shfl_xor(m, off, 32));
  if (lane == 0) red[wave] = m;
  __syncthreads();
  float M = red[0];
  for (int w = 1; w < 8; ++w) M = fmaxf(M, red[w]);
  __syncthreads();   // allow red[] reuse

  float e = __expf(s - M);
  eBuf[n] = e;
  float z = e;
  for (int off = 16; off > 0; off >>= 1) z += __shfl_xor(z, off, 32);
  if (lane == 0) red[wave] = z;
  __syncthreads();   // also publishes eBuf/tile for phase C
  float Z = 0.f;
  for (int w = 0; w < 8; ++w) Z += red[w];

  // Partial weighted sum via WMMA: p[d] = sum_n e[n] * kb[n,d].
  // Wave w owns d-tile [16w, 16w+16). A = e replicated across rows (16x4),
  // B = kb tile (4x16), accumulate over n in steps of 4.
  // ISA layout: VGPR0 -> K=0 (lanes 0-15) / K=2 (lanes 16-31); VGPR1 -> K=1/3.
  const int d0  = wave * 16;
  const int col = lane & 15;
  const int hi  = (lane >> 4) & 1;
  v8f c = {};
  for (int n0 = 0; n0 < CHUNK_N; n0 += 4) {
    v2f a, bm;
    a.x  = eBuf[n0 + 2 * hi];
    a.y  = eBuf[n0 + 2 * hi + 1];
    bm.x = tile[(n0 + 2 * hi)     * D + d0 + col];
    bm.y = tile[(n0 + 2 * hi + 1) * D + d0 + col];
    c = __builtin_amdgcn_wmma_f32_16x16x4_f32(
            /*neg_a=*/false, a, /*neg_b=*/false, bm,
            /*c_mod=*/(short)0, c, /*reuse_a=*/false, /*reuse_b=*/false);
  }

  // All D rows are identical; row M=0 lives in c[0], lanes 0-15 (N = col).
  const size_t pc = (size_t)(b * CHUNKS + chunk);
  if (t == 0) { ws_m[pc] = M; ws_z[pc] = Z; }
  if (hi == 0) ws_p[pc * D + d0 + col] = c[0];
}

// ---------------------------------------------------------------------------
// Kernel 2: combine chunk partials per batch (max-rescale), write output.
// ---------------------------------------------------------------------------
__global__ __launch_bounds__(128)
void combine_kernel(const float* __restrict__ ws_m,
                    const float* __restrict__ ws_z,
                    const float* __restrict__ ws_p,
                    float* __restrict__ out) {
  const int b = blockIdx.x;
  const int d = threadIdx.x;
  float M = NEG_INF;
  for (int c = 0; c < CHUNKS; ++c) M = fmaxf(M, ws_m[b * CHUNKS + c]);
  float Z = 0.f, x = 0.f;
  for (int c = 0; c < CHUNKS; ++c) {
    float r = __expf(ws_m[b * CHUNKS + c] - M);
    Z += r * ws_z[b * CHUNKS + c];
    x += r * ws_p[(size_t)(b * CHUNKS + c) * D + d];
  }
  out[(size_t)b * D + d] = x / Z;
}

// ---------------------------------------------------------------------------
extern "C" void kernel_launch(void* const* d_in, const int* in_sizes, int n_in,
                              void* d_out, int out_size, void* d_ws, size_t ws_size,
                              hipStream_t stream) {
  (void)in_sizes; (void)n_in; (void)out_size; (void)ws_size;
  const float* kb     = (const float*)d_in[0];
  const float* vecQ   = (const float*)d_in[1];
  const int*   objNum = (const int*)  d_in[2];
  const float* Wq     = (const float*)d_in[3];
  const float* bq     = (const float*)d_in[4];
  const float* Wa     = (const float*)d_in[5];
  const float* ba     = (const float*)d_in[6];
  float* out = (float*)d_out;
  float* ws  = (float*)d_ws;

  float* proj = ws + PROJ_OFF;
  float* wm   = ws + M_OFF;
  float* wz   = ws + Z_OFF;
  float* wp   = ws + P_OFF;

  proj_kernel<<<dim3(B_), dim3(D), 0, stream>>>(vecQ, Wq, bq, proj);
  att_partial_kernel<<<dim3(CHUNKS, B_), dim3(256), 0, stream>>>(
      kb, objNum, Wa, ba, proj, wm, wz, wp);
  combine_kernel<<<dim3(B_), dim3(D), 0, stream>>>(wm, wz, wp, out);
}